// ClassAttention_42812234006692
// MI455X (gfx1250) — compile-verified
//
#include <hip/hip_runtime.h>
#include <math.h>
#include <stdint.h>

// ---------------------------------------------------------------------------
// Class attention, restructured (single query row per head):
//   logit[b,h,n] = x[b,n,:] . w_eff[b,h,:] + bias[b,h]
//   xa[b,h,c]    = sum_n attn[b,h,n] * x[b,n,c]
//   x_cls        = (xa @ v_w + v_b) @ proj_w + proj_b
// Memory-bound: 2 passes over x (620 MB) ~= 27 us @ 23.3 TB/s.
// fp32 WMMA (v_wmma_f32_16x16x4_f32) for both heavy passes; x staged via
// CDNA5 async-to-LDS (ASYNCcnt) with double buffering to overlap HBM latency.
// ---------------------------------------------------------------------------

constexpr int Bn  = 32;
constexpr int Nn  = 3137;
constexpr int Cn  = 768;
constexpr int Hn  = 12;
constexpr int HDn = 64;
constexpr int ROWS = 64;       // rows of x per workgroup chunk
constexpr int CB   = 128;      // c-block staged in LDS at a time
constexpr int NCB  = Cn / CB;  // 6
constexpr int XROW = CB + 4;   // 132: pad -> lane stride 4 banks, 8B aligned

typedef float v2f __attribute__((ext_vector_type(2)));
typedef float v8f __attribute__((ext_vector_type(8)));

// ---- CDNA5 async global->LDS copy (ASYNCcnt-tracked), 16B per lane --------
__device__ __forceinline__ void async_ld_b128(const float* g, float* l) {
  uint32_t lofs = (uint32_t)(uintptr_t)(__attribute__((address_space(3))) float*)l;
  asm volatile("global_load_async_to_lds_b128 %0, %1, off"
               :: "v"(lofs), "v"((uint64_t)(uintptr_t)g) : "memory");
}

template <int N>
__device__ __forceinline__ void wait_async() {
#if __has_builtin(__builtin_amdgcn_s_wait_asynccnt)
  __builtin_amdgcn_s_wait_asynccnt(N);
#else
  asm volatile("s_wait_asynccnt %0" :: "i"(N) : "memory");
#endif
}

// ---------------- K1a: q[b,c'] = (x[b,0,:] @ q_w + q_b) * 64^-0.5 ----------
__global__ void k_qproj(const float* __restrict__ x, const float* __restrict__ q_w,
                        const float* __restrict__ q_b, float* __restrict__ qout) {
  extern __shared__ float sm[];                 // x0 [32][768]
  const int t = threadIdx.x;                    // 256 threads
  const int c0 = blockIdx.x * 128;
  for (int i = t; i < Bn * Cn; i += 256) {
    int b = i / Cn, c = i % Cn;
    sm[i] = x[(size_t)b * Nn * Cn + c];
  }
  __syncthreads();
  const int col = c0 + (t & 127);
  for (int b = (t >> 7); b < Bn; b += 2) {
    float acc = q_b[col];
    const float* xr = &sm[b * Cn];
    for (int c = 0; c < Cn; ++c) acc = fmaf(xr[c], q_w[(size_t)c * Cn + col], acc);
    qout[b * Cn + col] = acc * 0.125f;          // HD^-0.5 = 1/8
  }
}

// ---------------- K1b: w_eff[b,h,c] = sum_d k_w[c,h*64+d]*q[b,h*64+d] -----
__global__ void k_weff(const float* __restrict__ q, const float* __restrict__ k_w,
                       const float* __restrict__ k_b, float* __restrict__ weff,
                       float* __restrict__ bias) {
  __shared__ float kw[64][65];                  // pad 65: bank spread
  __shared__ float qh[32][64];
  const int t  = threadIdx.x;                   // 256
  const int ct = blockIdx.x;                    // 12 c-tiles of 64
  const int h  = blockIdx.y;                    // 12 heads
  const int c0 = ct * 64;
  for (int i = t; i < 64 * 64; i += 256) {
    int c = i >> 6, d = i & 63;
    kw[c][d] = k_w[(size_t)(c0 + c) * Cn + h * HDn + d];
  }
  for (int i = t; i < 32 * 64; i += 256) {
    int b = i >> 6, d = i & 63;
    qh[b][d] = q[b * Cn + h * HDn + d];
  }
  __syncthreads();
  const int cl = t & 63;
  for (int b = (t >> 6); b < Bn; b += 4) {
    float acc = 0.f;
    for (int d = 0; d < HDn; ++d) acc = fmaf(kw[cl][d], qh[b][d], acc);
    weff[((size_t)b * Hn + h) * Cn + c0 + cl] = acc;
  }
  if (ct == 0 && t < Bn) {
    float acc = 0.f;
    for (int d = 0; d < HDn; ++d) acc = fmaf(qh[t][d], k_b[h * HDn + d], acc);
    bias[t * Hn + h] = acc;
  }
}

// ---------------- K2: logits via WMMA f32 16x16x4 (pass 1 over x) ----------
// Per WG (chunk, b): 4 waves, each owns a PRIVATE double-buffered 16x128 x
// tile filled with global_load_async_to_lds_b128 -> no intra-loop barriers,
// only per-wave s_wait_asynccnt. D[h=M][n=N], K over c.
__global__ void k_logits(const float* __restrict__ x, const float* __restrict__ weff,
                         const float* __restrict__ bias, float* __restrict__ logits) {
  extern __shared__ float sm[];
  float* weff_s = sm;                           // [16][772]
  float* xb     = sm + 16 * 772;                // [4 waves][2 buf][16][XROW]
  float* bias_s = xb + 4 * 2 * 16 * XROW;       // [16]
  const int t    = threadIdx.x;                 // 128 threads = 4 waves
  const int lane = t & 31, wv = t >> 5;
  const int b     = blockIdx.y;
  const int chunk = blockIdx.x;
  for (int i = t; i < 16 * 772; i += 128) {
    int h = i / 772, c = i % 772;
    weff_s[i] = (h < Hn && c < Cn) ? weff[((size_t)b * Hn + h) * Cn + c] : 0.f;
  }
  if (t < 16) bias_s[t] = (t < Hn) ? bias[b * Hn + t] : 0.f;
  __syncthreads();

  const int n0   = chunk * ROWS + wv * 16;
  const int half = lane >> 4, l16 = lane & 15;
  const int j    = lane * 4;                    // 32 lanes x 16B = one 128-f row
  float* myx = xb + wv * 2 * 16 * XROW;

  // Issue one async B128 per row (16 per wave per c-block). Rows past Nn are
  // address-clamped; their garbage only reaches D columns that are not stored.
  auto stage = [&](int cb, int p) {
    float* dst = myx + p * 16 * XROW;
#pragma unroll
    for (int r = 0; r < 16; ++r) {
      int n = n0 + r; if (n >= Nn) n = Nn - 1;
      async_ld_b128(x + ((size_t)b * Nn + n) * Cn + cb * CB + j,
                    dst + r * XROW + j);
    }
  };

  stage(0, 0);
  v8f D = {};
  for (int cb = 0; cb < NCB; ++cb) {
    if (cb + 1 < NCB) { stage(cb + 1, (cb + 1) & 1); wait_async<16>(); }
    else              { wait_async<0>(); }
    const float* bufp = myx + (cb & 1) * 16 * XROW;
    for (int k4 = 0; k4 < CB / 4; ++k4) {
      const int ca = 4 * k4 + 2 * half;
      v2f a  = *(const v2f*)&weff_s[l16 * 772 + cb * CB + ca];
      v2f bb = *(const v2f*)&bufp[l16 * XROW + ca];
      D = __builtin_amdgcn_wmma_f32_16x16x4_f32(false, a, false, bb,
                                                (short)0, D, false, false);
    }
  }
  const int n = n0 + l16;                       // D: N = lane%16, M = r + 8*half
  if (n < Nn) {
#pragma unroll
    for (int r = 0; r < 8; ++r) {
      const int h = r + 8 * half;
      if (h < Hn)
        logits[((size_t)b * Hn + h) * Nn + n] = D[r] + bias_s[h];
    }
  }
}

// ---------------- K3: softmax per (b,h); writes attn output, zeroes xa -----
__global__ void k_softmax(const float* __restrict__ logits, float* __restrict__ attn,
                          float* __restrict__ xa) {
  __shared__ float red[256];
  const int t  = threadIdx.x;                   // 256
  const int bh = blockIdx.x;                    // B*H
  const float* L = logits + (size_t)bh * Nn;
  float m = -INFINITY;
  for (int n = t; n < Nn; n += 256) m = fmaxf(m, L[n]);
  red[t] = m; __syncthreads();
  for (int s = 128; s > 0; s >>= 1) { if (t < s) red[t] = fmaxf(red[t], red[t + s]); __syncthreads(); }
  m = red[0]; __syncthreads();
  float sum = 0.f;
  for (int n = t; n < Nn; n += 256) sum += expf(L[n] - m);
  red[t] = sum; __syncthreads();
  for (int s = 128; s > 0; s >>= 1) { if (t < s) red[t] += red[t + s]; __syncthreads(); }
  const float inv = 1.0f / red[0];
  float* A = attn + (size_t)bh * Nn;
  for (int n = t; n < Nn; n += 256) A[n] = expf(L[n] - m) * inv;
  for (int c = t; c < Cn; c += 256) xa[(size_t)bh * Cn + c] = 0.f;  // init for K4 atomics
}

// ---------------- K4: xa[b,h,c] += attn . x via WMMA (pass 2 over x) -------
// Shared double-buffered 64x128 x tile via async-to-LDS; D[h=M][c=N], K = n.
// Rows past Nn are address-clamped; attn weight 0 annihilates them.
__global__ void k_av(const float* __restrict__ x, const float* __restrict__ attn,
                     float* __restrict__ xa) {
  extern __shared__ float sm[];
  float* xb = sm;                               // [2 buf][64][XROW]
  float* at = sm + 2 * 64 * XROW;               // [16][68]
  const int t    = threadIdx.x;                 // 128 = 4 waves
  const int lane = t & 31, wv = t >> 5;
  const int b = blockIdx.y, chunk = blockIdx.x;
  const int nbase = chunk * ROWS;
  for (int i = t; i < 16 * 68; i += 128) {
    int h = i / 68, nn = i % 68;
    int n = nbase + nn;
    at[i] = (h < Hn && nn < ROWS && n < Nn)
                ? attn[((size_t)b * Hn + h) * Nn + n] : 0.f;
  }
  const int half = lane >> 4, l16 = lane & 15;

  auto stage = [&](int cb, int p) {             // 2048 B128s / 128 threads
    float* dst = xb + p * 64 * XROW;
#pragma unroll
    for (int i = 0; i < 16; ++i) {
      int flat4 = t + 128 * i;
      int r = flat4 >> 5, jj = (flat4 & 31) * 4;
      int n = nbase + r; if (n >= Nn) n = Nn - 1;
      async_ld_b128(x + ((size_t)b * Nn + n) * Cn + cb * CB + jj,
                    dst + r * XROW + jj);
    }
  };

  stage(0, 0);
  for (int cb = 0; cb < NCB; ++cb) {
    if (cb + 1 < NCB) { stage(cb + 1, (cb + 1) & 1); wait_async<16>(); }
    else              { wait_async<0>(); }
    __syncthreads();                            // all waves' loads resident
    const float* bufp = xb + (cb & 1) * 64 * XROW;
    for (int tl = 0; tl < 2; ++tl) {            // 8 c-tiles / 4 waves
      const int cl0 = (wv * 2 + tl) * 16;
      v8f D = {};
      for (int nb = 0; nb < ROWS / 4; ++nb) {
        v2f a = *(const v2f*)&at[l16 * 68 + 4 * nb + 2 * half];
        v2f bb;
        bb.x = bufp[(4 * nb + 2 * half)     * XROW + cl0 + l16];
        bb.y = bufp[(4 * nb + 2 * half + 1) * XROW + cl0 + l16];
        D = __builtin_amdgcn_wmma_f32_16x16x4_f32(false, a, false, bb,
                                                  (short)0, D, false, false);
      }
      const int cg = cb * CB + cl0 + l16;
#pragma unroll
      for (int r = 0; r < 8; ++r) {
        const int h = r + 8 * half;
        if (h < Hn)
          atomicAdd(&xa[((size_t)b * Hn + h) * Cn + cg], D[r]);  // coalesced
      }
    }
    __syncthreads();                            // reads done before rewrite
  }
}

// ---------------- K5: pre[b, h*64+cc] = xa[b,h,:] @ v_w(:, col) + v_b ------
__global__ void k_vproj(const float* __restrict__ xa, const float* __restrict__ v_w,
                        const float* __restrict__ v_b, float* __restrict__ pre) {
  extern __shared__ float sm[];                 // xa_h [32][768]
  const int t = threadIdx.x;                    // 256
  const int h = blockIdx.x;                     // 12
  for (int i = t; i < Bn * Cn; i += 256) {
    int b = i / Cn, c = i % Cn;
    sm[i] = xa[((size_t)b * Hn + h) * Cn + c];
  }
  __syncthreads();
  const int col = h * HDn + (t & 63);
  for (int b = (t >> 6); b < Bn; b += 4) {
    float acc = v_b[col];
    const float* xr = &sm[b * Cn];
    for (int c = 0; c < Cn; ++c) acc = fmaf(xr[c], v_w[(size_t)c * Cn + col], acc);
    pre[b * Cn + col] = acc;
  }
}

// ---------------- K6: x_cls = pre @ proj_w + proj_b ------------------------
__global__ void k_proj(const float* __restrict__ pre, const float* __restrict__ proj_w,
                       const float* __restrict__ proj_b, float* __restrict__ out) {
  extern __shared__ float sm[];                 // pre [32][768]
  const int t = threadIdx.x;                    // 256
  const int c0 = blockIdx.x * 128;
  for (int i = t; i < Bn * Cn; i += 256) sm[i] = pre[i];
  __syncthreads();
  const int col = c0 + (t & 127);
  for (int b = (t >> 7); b < Bn; b += 2) {
    float acc = proj_b[col];
    const float* xr = &sm[b * Cn];
    for (int c = 0; c < Cn; ++c) acc = fmaf(xr[c], proj_w[(size_t)c * Cn + col], acc);
    out[b * Cn + col] = acc;
  }
}

// ---------------------------------------------------------------------------
extern "C" void kernel_launch(void* const* d_in, const int* in_sizes, int n_in,
                              void* d_out, int out_size, void* d_ws, size_t ws_size,
                              hipStream_t stream) {
  (void)in_sizes; (void)n_in; (void)out_size; (void)ws_size;
  const float* x   = (const float*)d_in[0];
  const float* q_w = (const float*)d_in[1];
  const float* q_b = (const float*)d_in[2];
  const float* k_w = (const float*)d_in[3];
  const float* k_b = (const float*)d_in[4];
  const float* v_w = (const float*)d_in[5];
  const float* v_b = (const float*)d_in[6];
  const float* p_w = (const float*)d_in[7];
  const float* p_b = (const float*)d_in[8];

  float* out  = (float*)d_out;
  float* xcls = out;                            // [B,1,C]
  float* attn = out + Bn * Cn;                  // [B,H,1,N]

  float* ws   = (float*)d_ws;                   // ~7.0 MB total
  float* q    = ws;                                          // B*C
  float* weff = q    + (size_t)Bn * Cn;                      // B*H*C
  float* bias = weff + (size_t)Bn * Hn * Cn;                 // B*H
  float* logi = bias + (size_t)Bn * Hn;                      // B*H*N
  float* xa   = logi + (size_t)Bn * Hn * Nn;                 // B*H*C
  float* pre  = xa   + (size_t)Bn * Hn * Cn;                 // B*C

  const int chunks = (Nn + ROWS - 1) / ROWS;    // 50
  const size_t smx = (size_t)Bn * Cn * sizeof(float);                    // 96 KB
  const size_t sm2 = (size_t)(16 * 772 + 4 * 2 * 16 * XROW + 16) * 4;    // ~114 KB
  const size_t sm4 = (size_t)(2 * 64 * XROW + 16 * 68) * 4;              // ~70 KB

  k_qproj  <<<dim3(Cn / 128),    dim3(256), smx, stream>>>(x, q_w, q_b, q);
  k_weff   <<<dim3(Cn / 64, Hn), dim3(256), 0,   stream>>>(q, k_w, k_b, weff, bias);
  k_logits <<<dim3(chunks, Bn),  dim3(128), sm2, stream>>>(x, weff, bias, logi);
  k_softmax<<<dim3(Bn * Hn),     dim3(256), 0,   stream>>>(logi, attn, xa);
  k_av     <<<dim3(chunks, Bn),  dim3(128), sm4, stream>>>(x, attn, xa);
  k_vproj  <<<dim3(Hn),          dim3(256), smx, stream>>>(xa, v_w, v_b, pre);
  k_proj   <<<dim3(Cn / 128),    dim3(256), smx, stream>>>(pre, p_w, p_b, xcls);
}